// RPN_47029891891461
// MI455X (gfx1250) — compile-verified
//
#include <hip/hip_runtime.h>
#include <stdint.h>

// ---------------- RPN config (must match reference) ----------------
#define NIMG   8
#define NANCH  250000
#define KPRE   2000
#define KPOST  1000
#define CAP    4096          // candidate capacity for the bitonic sort
#define WORDS  63            // ceil(KPRE/32)
#define NMS_T  0.7f
#define IMGW   1333.0f
#define IMGH   800.0f
#define XFCLIP 4.135166556742356f
#define TORM   1.0f
#define PAIRS_PER_THREAD 2   // k_mask: (row,word) pairs per thread

#if defined(__has_builtin)
# if __has_builtin(__builtin_amdgcn_tensor_load_to_lds)
#  define HAVE_TDM 1
# endif
#endif

typedef unsigned uint4v __attribute__((ext_vector_type(4)));
typedef int      int4v  __attribute__((ext_vector_type(4)));
typedef int      int8v  __attribute__((ext_vector_type(8)));

// monotonic float->uint key (descending float == descending key)
__device__ __forceinline__ unsigned ordkey(float f) {
  unsigned b = __float_as_uint(f);
  return (b & 0x80000000u) ? ~b : (b | 0x80000000u);
}

// ---------------- 0: zero scratch ----------------
__global__ void k_zero(unsigned* __restrict__ h1, unsigned* __restrict__ h2,
                       unsigned* __restrict__ cc) {
  const size_t n = (size_t)NIMG * 65536;
  for (size_t i = (size_t)blockIdx.x * blockDim.x + threadIdx.x; i < n;
       i += (size_t)gridDim.x * blockDim.x) {
    h1[i] = 0u; h2[i] = 0u;
  }
  if (blockIdx.x == 0 && threadIdx.x < NIMG) cc[threadIdx.x] = 0u;
}

// ---------------- 1: histogram of top-16 key bits ----------------
__global__ void k_hist1(const float* __restrict__ obj, unsigned* __restrict__ h1) {
  const int img = blockIdx.y;
  const int i = blockIdx.x * blockDim.x + threadIdx.x;
  if (i < NANCH) {
    const float* p = obj + (size_t)img * NANCH + i;
    __builtin_prefetch(p + 8192, 0, 0);           // global_prefetch_b8
    unsigned k = ordkey(*p);
    atomicAdd(&h1[img * 65536 + (k >> 16)], 1u);
  }
}

// ---------------- 2: find msb bin holding the 2000th element (parallel) ----------------
__global__ void __launch_bounds__(256)
k_findbin(const unsigned* __restrict__ h1, unsigned* __restrict__ tinfo) {
  __shared__ unsigned psum[256];
  __shared__ unsigned suf[256];
  const int img = blockIdx.x, t = threadIdx.x;
  const unsigned* h = h1 + img * 65536;
  const int b0 = t * 256;
  unsigned s = 0;
  for (int k = 0; k < 256; ++k) s += h[b0 + k];
  psum[t] = s;
  __syncthreads();
  if (t == 0) {
    unsigned run = 0;
    for (int c = 255; c >= 0; --c) { suf[c] = run; run += psum[c]; }
  }
  __syncthreads();
  const unsigned above = suf[t];
  if (above < KPRE && above + psum[t] >= KPRE) {    // unique owning chunk
    unsigned cum = above;
    for (int b = b0 + 255; b >= b0; --b) {
      unsigned c = h[b];
      if (cum + c >= KPRE) {
        tinfo[img * 4 + 0] = (unsigned)b;           // msb bin
        tinfo[img * 4 + 1] = cum;                   // count strictly above bin b
        break;
      }
      cum += c;
    }
  }
}

// ---------------- 3: histogram of low-16 bits inside threshold bin ----------------
__global__ void k_hist2(const float* __restrict__ obj, const unsigned* __restrict__ tinfo,
                        unsigned* __restrict__ h2) {
  const int img = blockIdx.y;
  const int i = blockIdx.x * blockDim.x + threadIdx.x;
  if (i < NANCH) {
    unsigned k = ordkey(obj[(size_t)img * NANCH + i]);
    if ((k >> 16) == tinfo[img * 4 + 0])
      atomicAdd(&h2[img * 65536 + (k & 0xFFFFu)], 1u);
  }
}

// ---------------- 4: exact 2000th-largest key threshold (parallel) ----------------
__global__ void __launch_bounds__(256)
k_findthr(const unsigned* __restrict__ h2, unsigned* __restrict__ tinfo) {
  __shared__ unsigned psum[256];
  __shared__ unsigned suf[256];
  const int img = blockIdx.x, t = threadIdx.x;
  const unsigned bhi = tinfo[img * 4 + 0];
  const unsigned C1  = tinfo[img * 4 + 1];
  const unsigned* h = h2 + img * 65536;
  const int b0 = t * 256;
  unsigned s = 0;
  for (int k = 0; k < 256; ++k) s += h[b0 + k];
  psum[t] = s;
  __syncthreads();
  if (t == 0) {
    unsigned run = 0;
    for (int c = 255; c >= 0; --c) { suf[c] = run; run += psum[c]; }
  }
  __syncthreads();
  const unsigned above = suf[t] + C1;
  if (above < KPRE && above + psum[t] >= KPRE) {
    unsigned cum = above;
    for (int b = b0 + 255; b >= b0; --b) {
      unsigned c = h[b];
      if (cum + c >= KPRE) {
        tinfo[img * 4 + 2] = (bhi << 16) | (unsigned)b;  // exact 2000th-largest key
        break;
      }
      cum += c;
    }
  }
}

// ---------------- 5: compact key >= T ----------------
__global__ void k_compact(const float* __restrict__ obj, const unsigned* __restrict__ tinfo,
                          unsigned* __restrict__ cc, unsigned* __restrict__ ckeys,
                          unsigned* __restrict__ cidx) {
  const int img = blockIdx.y;
  const int i = blockIdx.x * blockDim.x + threadIdx.x;
  if (i < NANCH) {
    unsigned k = ordkey(obj[(size_t)img * NANCH + i]);
    if (k >= tinfo[img * 4 + 2]) {
      unsigned p = atomicAdd(&cc[img], 1u);
      if (p < CAP) { ckeys[img * CAP + p] = k; cidx[img * CAP + p] = (unsigned)i; }
    }
  }
}

// ---------------- 6: bitonic sort (key desc, index asc), take top-2000 ----------------
__global__ void __launch_bounds__(1024)
k_sort(const unsigned* __restrict__ cc, const unsigned* __restrict__ ckeys,
       const unsigned* __restrict__ cidx, unsigned* __restrict__ sidx) {
  __shared__ unsigned long long sk[CAP];           // 32 KB
  const int img = blockIdx.x;
  unsigned n = cc[img]; if (n > CAP) n = CAP;
  for (int t = threadIdx.x; t < CAP; t += 1024) {
    if ((unsigned)t < n)
      sk[t] = ((unsigned long long)ckeys[img * CAP + t] << 32) |
              (unsigned)(~cidx[img * CAP + t]);
    else
      sk[t] = 0ull;
  }
  __syncthreads();
  for (int k = 2; k <= CAP; k <<= 1) {
    for (int j = k >> 1; j > 0; j >>= 1) {
      for (int t = threadIdx.x; t < CAP; t += 1024) {
        int ixj = t ^ j;
        if (ixj > t) {
          unsigned long long a = sk[t], b = sk[ixj];
          bool desc = ((t & k) == 0);
          if (desc ? (a < b) : (a > b)) { sk[t] = b; sk[ixj] = a; }
        }
      }
      __syncthreads();
    }
  }
  for (int r = threadIdx.x; r < KPRE; r += 1024) {
    unsigned long long v = sk[r];
    sidx[img * KPRE + r] = ~(unsigned)(v & 0xFFFFFFFFull);
  }
}

// ---------------- 7: decode + clip + validity for selected boxes ----------------
__global__ void k_decode(const float* __restrict__ anchors, const float* __restrict__ obj,
                         const float* __restrict__ deltas, const unsigned* __restrict__ sidx,
                         float* __restrict__ props, float* __restrict__ scores,
                         unsigned* __restrict__ validArr) {
  const int gid = blockIdx.x * blockDim.x + threadIdx.x;
  if (gid >= NIMG * KPRE) return;
  const int img = gid / KPRE;
  unsigned idx = sidx[gid];
  if (idx >= NANCH) idx = NANCH - 1;             // pad safety
  const float* a = anchors + (size_t)idx * 4;
  float w  = a[2] - a[0] + TORM;
  float h  = a[3] - a[1] + TORM;
  float cx = a[0] + 0.5f * w;
  float cy = a[1] + 0.5f * h;
  const float* d = deltas + ((size_t)img * NANCH + idx) * 4;
  float dx = d[0], dy = d[1];
  float dw = fminf(d[2], XFCLIP);
  float dh = fminf(d[3], XFCLIP);
  float pcx = dx * w + cx, pcy = dy * h + cy;
  float pw = __expf(dw) * w, ph = __expf(dh) * h;
  float x1 = pcx - 0.5f * pw;
  float y1 = pcy - 0.5f * ph;
  float x2 = pcx + 0.5f * pw - 1.0f;
  float y2 = pcy + 0.5f * ph - 1.0f;
  x1 = fminf(fmaxf(x1, 0.0f), IMGW - TORM);
  y1 = fminf(fmaxf(y1, 0.0f), IMGH - TORM);
  x2 = fminf(fmaxf(x2, 0.0f), IMGW - TORM);
  y2 = fminf(fmaxf(y2, 0.0f), IMGH - TORM);
  float* o = props + (size_t)gid * 4;
  o[0] = x1; o[1] = y1; o[2] = x2; o[3] = y2;
  scores[gid]   = obj[(size_t)img * NANCH + idx];
  validArr[gid] = (((x2 - x1 + TORM) >= 0.0f) && ((y2 - y1 + TORM) >= 0.0f)) ? 1u : 0u;
}

// ---------------- 8: pairwise IoU suppression bitmask ----------------
// 32 KB contiguous box tile staged global->LDS with the Tensor Data Mover (TENSORcnt);
// per-lane async-to-LDS fallback if the TDM builtin is unavailable.
__global__ void __launch_bounds__(256)
k_mask(const float* __restrict__ props, unsigned* __restrict__ mask) {
  __shared__ float sb[KPRE * 4];                  // 32 KB
  __shared__ float sa[KPRE];                      //  8 KB
  const int img = blockIdx.y;
  const float* base = props + (size_t)img * KPRE * 4;
#if defined(HAVE_TDM)
  if (threadIdx.x < 32) {                         // wave 0 issues the DMA
    const unsigned lds_base = (unsigned)(uintptr_t)&sb[0];
    const unsigned long long ga = (unsigned long long)(uintptr_t)base;
    const unsigned ELEMS = KPRE * 4;              // 8000 x 4-byte elements
    // D# group 0: count=1 | lds_addr | global_addr[56:0] | type=2 (bits 127:126)
    uint4v g0;
    g0.x = 1u;
    g0.y = lds_base;
    g0.z = (unsigned)ga;
    g0.w = ((unsigned)(ga >> 32) & 0x01FFFFFFu) | 0x80000000u;
    // D# group 1: data_size=2 (4B) | tensor_dim0=tile_dim0=8000 | 1-D tile | stride=8000
    int8v g1;
    g1[0] = 0x20000;                              // data_size code 2 at bits[17:16]
    g1[1] = (int)((ELEMS & 0xFFFFu) << 16);       // tensor_dim0[15:0]  -> bits 63:48
    g1[2] = (int)(ELEMS >> 16);                   // tensor_dim0[31:16] -> bits 79:64
    g1[3] = (int)((ELEMS & 0xFFFFu) << 16);       // tile_dim0          -> bits 127:112
    g1[4] = 0;                                    // tile_dim1=0, tile_dim2=0 (unused)
    g1[5] = (int)ELEMS;                           // tensor_dim0_stride[31:0]
    g1[6] = 0; g1[7] = 0;
    int4v z4 = {0, 0, 0, 0};
#if __clang_major__ >= 23
    int8v z8 = {0, 0, 0, 0, 0, 0, 0, 0};
    __builtin_amdgcn_tensor_load_to_lds(g0, g1, z4, z4, z8, 0);
#else
    __builtin_amdgcn_tensor_load_to_lds(g0, g1, z4, z4, 0);
#endif
    __builtin_amdgcn_s_wait_tensorcnt(0);
  }
  __syncthreads();
#else
  for (int r = threadIdx.x; r < KPRE; r += 256) {
    unsigned lds = (unsigned)(uintptr_t)&sb[r * 4];
    const float* g = base + (size_t)r * 4;
    asm volatile("global_load_async_to_lds_b128 %0, %1, off"
                 :: "v"(lds), "v"(g) : "memory");
  }
  asm volatile("s_wait_asynccnt 0" ::: "memory");
  __syncthreads();
#endif
  for (int r = threadIdx.x; r < KPRE; r += 256)
    sa[r] = (sb[r * 4 + 2] - sb[r * 4 + 0] + TORM) *
            (sb[r * 4 + 3] - sb[r * 4 + 1] + TORM);
  __syncthreads();
  #pragma unroll
  for (int pp = 0; pp < PAIRS_PER_THREAD; ++pp) {
    const int p = (blockIdx.x * PAIRS_PER_THREAD + pp) * 256 + threadIdx.x;
    if (p < KPRE * WORDS) {
      const int i = p / WORDS, w = p % WORDS;
      const float x1 = sb[i * 4], y1 = sb[i * 4 + 1];
      const float x2 = sb[i * 4 + 2], y2 = sb[i * 4 + 3];
      const float ai = sa[i];
      unsigned bits = 0u;
      const int j0 = w * 32;
      #pragma unroll 4
      for (int jj = 0; jj < 32; ++jj) {
        int j = j0 + jj;
        if (j < KPRE && j > i) {
          float xx1 = fmaxf(x1, sb[j * 4]);
          float yy1 = fmaxf(y1, sb[j * 4 + 1]);
          float xx2 = fminf(x2, sb[j * 4 + 2]);
          float yy2 = fminf(y2, sb[j * 4 + 3]);
          float iw = fmaxf(xx2 - xx1 + TORM, 0.0f);
          float ih = fmaxf(yy2 - yy1 + TORM, 0.0f);
          float inter = iw * ih;
          float iou = inter / (ai + sa[j] - inter);
          if (iou > NMS_T) bits |= 1u << jj;
        }
      }
      mask[((size_t)img * KPRE + i) * WORDS + w] = bits;
    }
  }
}

// ---------------- 9: greedy NMS reduction, single wave32, register-resident ----------------
__global__ void __launch_bounds__(32)
k_reduce(const unsigned* __restrict__ mask, const unsigned* __restrict__ validArr,
         unsigned* __restrict__ keep) {
  const int img = blockIdx.x, t = threadIdx.x;    // t in [0,32)
  unsigned v0 = 0u, v1 = 0u;
  for (int jj = 0; jj < 32; ++jj) {
    const int jA = t * 32 + jj;                   // word t: j < 1024, always valid
    if (validArr[img * KPRE + jA]) v0 |= 1u << jj;
    const int jB = (t + 32) * 32 + jj;            // word t+32
    if (jB < KPRE && (t + 32) < WORDS && validArr[img * KPRE + jB]) v1 |= 1u << jj;
  }
  unsigned rem0 = ~v0, rem1 = ~v1;
  const unsigned* m = mask + (size_t)img * KPRE * WORDS;
  for (int i = 0; i < KPRE; ++i) {
    const int w = i >> 5;
    const unsigned word = (w < 32) ? __shfl(rem0, w, 32) : __shfl(rem1, w - 32, 32);
    if (((word >> (i & 31)) & 1u) == 0u) {        // box i kept -> suppress its row
      rem0 |= m[(size_t)i * WORDS + t];
      if (t + 32 < WORDS) rem1 |= m[(size_t)i * WORDS + t + 32];
    }
  }
  keep[img * WORDS + t] = ~rem0;
  if (t + 32 < WORDS) {
    unsigned k1 = ~rem1;
    if (t + 32 == WORDS - 1) k1 &= (1u << (KPRE - 32 * (WORDS - 1))) - 1u; // bits >= 2000 off
    keep[img * WORDS + t + 32] = k1;
  }
}

// ---------------- 10: stable partition (== top_k over masked scores) ----------------
// props tile staged into LDS with CDNA5 async copy (ASYNCcnt)
__global__ void __launch_bounds__(256)
k_out(const unsigned* __restrict__ keep, const float* __restrict__ props,
      const float* __restrict__ scores, float* __restrict__ out) {
  __shared__ float sp[KPRE * 4];                  // 32 KB
  __shared__ unsigned kw[WORDS];
  __shared__ unsigned short pos[KPRE];
  const int img = blockIdx.x, t = threadIdx.x;
  for (int r = t; r < KPRE; r += 256) {
    unsigned lds = (unsigned)(uintptr_t)&sp[r * 4];
    const float* g = props + ((size_t)img * KPRE + r) * 4;
    asm volatile("global_load_async_to_lds_b128 %0, %1, off"
                 :: "v"(lds), "v"(g) : "memory");
  }
  if (t < WORDS) kw[t] = keep[img * WORDS + t];
  asm volatile("s_wait_asynccnt 0" ::: "memory");
  __syncthreads();
  if (t == 0) {
    int nk = 0;
    for (int r = 0; r < KPRE; ++r)
      if ((kw[r >> 5] >> (r & 31)) & 1u) pos[r] = (unsigned short)nk++;
    int np = nk;
    for (int r = 0; r < KPRE; ++r)
      if (!((kw[r >> 5] >> (r & 31)) & 1u)) pos[r] = (unsigned short)np++;
  }
  __syncthreads();
  for (int r = t; r < KPRE; r += 256) {
    unsigned p = pos[r];
    if (p < KPOST) {
      bool kbit = (kw[r >> 5] >> (r & 31)) & 1u;
      float* o = out + ((size_t)img * KPOST + p) * 5;
      o[0] = sp[r * 4 + 0]; o[1] = sp[r * 4 + 1];
      o[2] = sp[r * 4 + 2]; o[3] = sp[r * 4 + 3];
      o[4] = kbit ? scores[img * KPRE + r] : -1e9f;
    }
  }
}

// ---------------- launcher ----------------
extern "C" void kernel_launch(void* const* d_in, const int* in_sizes, int n_in,
                              void* d_out, int out_size, void* d_ws, size_t ws_size,
                              hipStream_t stream) {
  (void)in_sizes; (void)n_in; (void)out_size; (void)ws_size;
  const float* anchors = (const float*)d_in[0];   // [250000,4]
  const float* obj     = (const float*)d_in[1];   // [8,250000]
  const float* deltas  = (const float*)d_in[2];   // [8,250000,4]
  float* out = (float*)d_out;                     // [8,1000,5]

  char* ws = (char*)d_ws;
  size_t off = 0;
  auto alloc = [&](size_t bytes) { char* p = ws + off; off = (off + bytes + 255) & ~(size_t)255; return p; };
  unsigned* h1     = (unsigned*)alloc((size_t)NIMG * 65536 * 4);
  unsigned* h2     = (unsigned*)alloc((size_t)NIMG * 65536 * 4);
  unsigned* tinfo  = (unsigned*)alloc((size_t)NIMG * 4 * 4);
  unsigned* cc     = (unsigned*)alloc((size_t)NIMG * 4);
  unsigned* ckeys  = (unsigned*)alloc((size_t)NIMG * CAP * 4);
  unsigned* cidx   = (unsigned*)alloc((size_t)NIMG * CAP * 4);
  unsigned* sidx   = (unsigned*)alloc((size_t)NIMG * KPRE * 4);
  float*    props  = (float*)   alloc((size_t)NIMG * KPRE * 4 * 4);
  float*    scores = (float*)   alloc((size_t)NIMG * KPRE * 4);
  unsigned* validA = (unsigned*)alloc((size_t)NIMG * KPRE * 4);
  unsigned* mask   = (unsigned*)alloc((size_t)NIMG * KPRE * WORDS * 4);
  unsigned* keep   = (unsigned*)alloc((size_t)NIMG * WORDS * 4);

  const dim3 gimg((NANCH + 255) / 256, NIMG);
  const int pairs = KPRE * WORDS;
  const int mask_blocks = (pairs + 256 * PAIRS_PER_THREAD - 1) / (256 * PAIRS_PER_THREAD);

  k_zero   <<<512, 256, 0, stream>>>(h1, h2, cc);
  k_hist1  <<<gimg, 256, 0, stream>>>(obj, h1);
  k_findbin<<<NIMG, 256, 0, stream>>>(h1, tinfo);
  k_hist2  <<<gimg, 256, 0, stream>>>(obj, tinfo, h2);
  k_findthr<<<NIMG, 256, 0, stream>>>(h2, tinfo);
  k_compact<<<gimg, 256, 0, stream>>>(obj, tinfo, cc, ckeys, cidx);
  k_sort   <<<NIMG, 1024, 0, stream>>>(cc, ckeys, cidx, sidx);
  k_decode <<<(NIMG * KPRE + 255) / 256, 256, 0, stream>>>(anchors, obj, deltas, sidx,
                                                           props, scores, validA);
  k_mask   <<<dim3(mask_blocks, NIMG), 256, 0, stream>>>(props, mask);
  k_reduce <<<NIMG, 32, 0, stream>>>(mask, validA, keep);
  k_out    <<<NIMG, 256, 0, stream>>>(keep, props, scores, out);
}